// UnifiedPresenceLoss_48782238548159
// MI455X (gfx1250) — compile-verified
//
#include <hip/hip_runtime.h>

// ---------------------------------------------------------------------------
// UnifiedPresenceLoss for MI455X (gfx1250, wave32)
//   stage 1: bandwidth-bound streaming of logits/labels (float4 loads),
//            per-thread masked-BCE accumulation, wave32 shuffle + LDS reduce,
//            one (loss,mask) partial pair per block -> d_ws
//   stage 2: single wave; 512 partials per array reduced with chained
//            V_WMMA_F32_16X16X4_F32 (A = ones) accumulations, then a 32-lane
//            butterfly; lane 0 writes loss_sum / max(mask_sum, 1).
// ---------------------------------------------------------------------------

typedef __attribute__((ext_vector_type(2))) float v2f;
typedef __attribute__((ext_vector_type(8))) float v8f;

#define UPL_BATCH      65536
#define UPL_UNIFIED    256
#define UPL_SLOTS      64
#define UPL_TOOLS      128
#define UPL_PARAMS     192     // UPL_UNIFIED - UPL_SLOTS
#define UPL_NBLOCKS    512
#define UPL_THREADS    256

__global__ void upl_stage1(const float4* __restrict__ logits4,   // [B*64] float4
                           const float4* __restrict__ labels4,   // [B*64] float4
                           const int*    __restrict__ decision,  // [B]
                           const int*    __restrict__ tool,      // [B]
                           const float*  __restrict__ tpm,       // [128*192]
                           float*        __restrict__ ws)        // [2*NBLOCKS]
{
    const int tid     = threadIdx.x;
    const int rowlane = tid & 63;   // which float4 (4 columns) within a row
    const int rowsel  = tid >> 6;   // 0..3: row within the 4-row group

    float loss_sum = 0.0f;
    float mask_sum = 0.0f;

    // 65536 rows / 4 rows-per-group = 16384 groups; 512 blocks -> 32 iters each
    for (int grp = blockIdx.x; grp < (UPL_BATCH / 4); grp += gridDim.x) {
        const int  row  = grp * 4 + rowsel;
        const int  t    = tool[row];
        const bool cond = (decision[row] == 1) && (t >= 0) && (t < UPL_TOOLS);

        const float4 x4 = logits4[row * 64 + rowlane];
        const float4 y4 = labels4[row * 64 + rowlane];

        float m0, m1, m2, m3;
        if (rowlane < (UPL_SLOTS / 4)) {
            // slot columns: always valid
            m0 = m1 = m2 = m3 = 1.0f;
        } else if (cond) {
            const int base = t * UPL_PARAMS + (rowlane * 4 - UPL_SLOTS);
            m0 = tpm[base + 0]; m1 = tpm[base + 1];
            m2 = tpm[base + 2]; m3 = tpm[base + 3];
        } else {
            m0 = m1 = m2 = m3 = 0.0f;
        }

        const float xs[4] = {x4.x, x4.y, x4.z, x4.w};
        const float ys[4] = {y4.x, y4.y, y4.z, y4.w};
        const float ms[4] = {m0, m1, m2, m3};

        #pragma unroll
        for (int k = 0; k < 4; ++k) {
            const float x = xs[k];
            const float y = ys[k];
            // numerically stable BCE-with-logits
            const float l = fmaxf(x, 0.0f) - x * y + log1pf(__expf(-fabsf(x)));
            loss_sum = fmaf(l, ms[k], loss_sum);
            mask_sum += ms[k];
        }
    }

    // wave32 reduction
    #pragma unroll
    for (int off = 16; off > 0; off >>= 1) {
        loss_sum += __shfl_down(loss_sum, off, 32);
        mask_sum += __shfl_down(mask_sum, off, 32);
    }

    __shared__ float sl[UPL_THREADS / 32];
    __shared__ float sm[UPL_THREADS / 32];
    const int wave = tid >> 5;
    const int lane = tid & 31;
    if (lane == 0) { sl[wave] = loss_sum; sm[wave] = mask_sum; }
    __syncthreads();

    if (tid == 0) {
        float L = 0.0f, M = 0.0f;
        #pragma unroll
        for (int w = 0; w < UPL_THREADS / 32; ++w) { L += sl[w]; M += sm[w]; }
        ws[blockIdx.x]               = L;
        ws[UPL_NBLOCKS + blockIdx.x] = M;
    }
}

// One wave (32 threads), EXEC all-ones (WMMA requirement: no divergence
// before the matrix ops). Reduces 512 loss partials and 512 mask partials.
__global__ void upl_stage2(const float* __restrict__ ws, float* __restrict__ out)
{
    const int lane = threadIdx.x;           // 0..31

    v2f ones; ones.x = 1.0f; ones.y = 1.0f; // A = 16x4 all-ones
    v8f accL = {};                           // C/D accumulators (16x16 f32)
    v8f accM = {};

    // 512 values = 8 chunks of 64 (one v2f per lane per chunk).
    // D = ones(16x4) x B(4x16) + C  => every D element accumulates B
    // column-sums; exact operand->matrix-slot mapping is irrelevant for a
    // grand total since each value lands in exactly one B slot.
    #pragma unroll
    for (int j = 0; j < 8; ++j) {
        v2f bL, bM;
        bL.x = ws[lane + 32 * (2 * j + 0)];
        bL.y = ws[lane + 32 * (2 * j + 1)];
        bM.x = ws[UPL_NBLOCKS + lane + 32 * (2 * j + 0)];
        bM.y = ws[UPL_NBLOCKS + lane + 32 * (2 * j + 1)];

        accL = __builtin_amdgcn_wmma_f32_16x16x4_f32(
            /*neg_a=*/false, ones, /*neg_b=*/false, bL,
            /*c_mod=*/(short)0, accL, /*reuse_a=*/false, /*reuse_b=*/false);
        accM = __builtin_amdgcn_wmma_f32_16x16x4_f32(
            /*neg_a=*/false, ones, /*neg_b=*/false, bM,
            /*c_mod=*/(short)0, accM, /*reuse_a=*/false, /*reuse_b=*/false);
    }

    // VGPR0 of D: lanes 0-15 hold row M=0, lanes 16-31 hold row M=8.
    // All D rows are identical (A all-ones), so a full 32-lane butterfly of
    // element 0 yields 2 * total.
    float L = accL[0];
    float M = accM[0];
    #pragma unroll
    for (int off = 16; off > 0; off >>= 1) {
        L += __shfl_xor(L, off, 32);
        M += __shfl_xor(M, off, 32);
    }
    L *= 0.5f;
    M *= 0.5f;

    if (lane == 0) {
        out[0] = L / fmaxf(M, 1.0f);
    }
}

extern "C" void kernel_launch(void* const* d_in, const int* in_sizes, int n_in,
                              void* d_out, int out_size, void* d_ws, size_t ws_size,
                              hipStream_t stream) {
    const float4* logits4  = (const float4*)d_in[0];   // presence_logits [B,256] f32
    const float4* labels4  = (const float4*)d_in[1];   // presence_labels [B,256] f32
    const int*    decision = (const int*)d_in[2];      // decision_labels [B]
    const int*    tool     = (const int*)d_in[3];      // tool_name_labels [B]
    const float*  tpm      = (const float*)d_in[4];    // tool_param_mask [128,192] f32

    float* ws  = (float*)d_ws;                         // 2*512 floats = 4 KB
    float* out = (float*)d_out;

    upl_stage1<<<UPL_NBLOCKS, UPL_THREADS, 0, stream>>>(
        logits4, labels4, decision, tool, tpm, ws);
    upl_stage2<<<1, 32, 0, stream>>>(ws, out);
}